// GIN_44762149159634
// MI455X (gfx1250) — compile-verified
//
#include <hip/hip_runtime.h>

// ---------------------------------------------------------------------------
// GIN forward for MI455X (gfx1250, wave32, WMMA).
// Memory-bound by the edge scatter (agg fits in 192MB L2 -> L2-resident
// atomics); GEMMs use full-precision V_WMMA_F32_16X16X4_F32 tiles.
// ---------------------------------------------------------------------------

typedef float v2f __attribute__((ext_vector_type(2)));
typedef float v8f __attribute__((ext_vector_type(8)));

#define N_NODES  100000
#define N_GRAPHS 1024
#define DFEAT    128
#define BN_EPS   1e-5f

// ---------------------------------------------------------------- zero fill
__global__ void zero_f32(float* __restrict__ p, int n) {
  for (int i = blockIdx.x * blockDim.x + threadIdx.x; i < n;
       i += gridDim.x * blockDim.x)
    p[i] = 0.0f;
}

// ------------------------------------------------------------ edge scatter
// agg[dst] += x[src]; 32 lanes per edge, 4 floats per lane (float4 gather,
// 4 hardware f32 atomic adds -> global_atomic_add_f32, L2-resident target).
__global__ void edge_scatter(const float* __restrict__ x,
                             const int* __restrict__ src,
                             const int* __restrict__ dst,
                             float* __restrict__ agg, int n_edges) {
  int t = blockIdx.x * blockDim.x + threadIdx.x;
  int e = t >> 5;
  if (e >= n_edges) return;
  int lane = t & 31;
  int s = src[e];
  int d = dst[e];
  const float4 v = *(const float4*)(x + (size_t)s * DFEAT + lane * 4);
  float* out = agg + (size_t)d * DFEAT + lane * 4;
  unsafeAtomicAdd(out + 0, v.x);
  unsafeAtomicAdd(out + 1, v.y);
  unsafeAtomicAdd(out + 2, v.z);
  unsafeAtomicAdd(out + 3, v.w);
}

// ----------------------------------------------------------- global pooling
__global__ void pool_scatter(const float* __restrict__ x,
                             const int* __restrict__ batch,
                             float* __restrict__ g, int total) {
  int idx = blockIdx.x * blockDim.x + threadIdx.x;
  if (idx >= total) return;
  int node = idx >> 7;           // /128
  int col  = idx & 127;
  unsafeAtomicAdd(&g[(size_t)batch[node] * DFEAT + col], x[idx]);
}

// ------------------------------------------------------------- WMMA GEMM
// C[M x NOUT] = epilogue( pre(A)[M x 128] @ W[128 x NOUT] + bias )
//   PRE:  0 = identity, 1 = A + A2 (GIN x+agg), 2 = BatchNorm+ReLU via stats
//   POST: 0 = store, 1 = ReLU+store, 2 = store + per-column sum/sumsq atomics
// Block = (NOUT/16) waves; one 16-row strip per block, 16 cols per wave.
// A tile staged in LDS, row stride 132 to avoid bank conflicts.
//
// Fragment layouts (ISA 7.12.2, wave32):
//   A 16x4 f32:  lane l -> M=l&15; VGPR0 = K{0|2}, VGPR1 = K{1|3} by lane half
//   B 4x16 f32:  lane l -> N=l&15; VGPR0 = K{0|2}, VGPR1 = K{1|3} by lane half
//   C 16x16 f32: VGPR r -> M=r (lanes 0-15) / M=r+8 (lanes 16-31), N=l&15
template <int NOUT, int PRE, int POST>
__global__ void gemm_k128(const float* __restrict__ A,
                          const float* __restrict__ A2,
                          const float* __restrict__ W,
                          const float* __restrict__ bias,
                          float* __restrict__ C,
                          float* __restrict__ stats,   // [128] sum, [128] sumsq
                          const float* __restrict__ gamma,
                          const float* __restrict__ beta,
                          float invN) {
  __shared__ float As[16][132];

  const int tid   = threadIdx.x;
  const int mbase = blockIdx.x * 16;

  // Stage A tile (with fused pre-op) into LDS.
  for (int i = tid; i < 16 * DFEAT; i += blockDim.x) {
    int r = i >> 7;
    int c = i & 127;
    size_t off = (size_t)(mbase + r) * DFEAT + c;
    float v = A[off];
    if constexpr (PRE == 1) {
      v += A2[off];
    } else if constexpr (PRE == 2) {
      float mu  = stats[c] * invN;
      float var = stats[DFEAT + c] * invN - mu * mu;
      float inv = rsqrtf(var + BN_EPS);
      v = (v - mu) * inv * gamma[c] + beta[c];
      v = fmaxf(v, 0.0f);
    }
    As[r][c] = v;
  }
  __syncthreads();

  const int wave  = tid >> 5;
  const int lane  = tid & 31;
  const int lm    = lane & 15;
  const int khalf = (lane >> 4) << 1;   // 0 or 2
  const int nb    = wave * 16;          // column base of this wave's tile

  v8f c = {};
  #pragma unroll
  for (int k4 = 0; k4 < 32; ++k4) {
    const int k = (k4 << 2) + khalf;
    v2f a, b;
    a.x = As[lm][k];
    a.y = As[lm][k + 1];
    b.x = W[(size_t)k * NOUT + nb + lm];
    b.y = W[(size_t)(k + 1) * NOUT + nb + lm];
    c = __builtin_amdgcn_wmma_f32_16x16x4_f32(
        /*neg_a=*/false, a, /*neg_b=*/false, b,
        /*c_mod=*/(short)0, c, /*reuse_a=*/false, /*reuse_b=*/false);
  }

  const int col = nb + lm;
  const float bi = bias[col];
  float s = 0.0f, sq = 0.0f;
  #pragma unroll
  for (int r = 0; r < 8; ++r) {
    int row = mbase + r + ((lane >> 4) << 3);
    float v = c[r] + bi;
    if constexpr (POST == 1) v = fmaxf(v, 0.0f);
    C[(size_t)row * NOUT + col] = v;
    if constexpr (POST == 2) { s += v; sq += v * v; }
  }
  if constexpr (POST == 2) {
    unsafeAtomicAdd(&stats[col], s);
    unsafeAtomicAdd(&stats[DFEAT + col], sq);
  }
}

// ---------------------------------------------------------------------------
extern "C" void kernel_launch(void* const* d_in, const int* in_sizes, int n_in,
                              void* d_out, int out_size, void* d_ws,
                              size_t ws_size, hipStream_t stream) {
  const float* x     = (const float*)d_in[0];
  const int*   ei    = (const int*)d_in[1];     // [2, E]
  const int*   batch = (const int*)d_in[2];
  const float* W1    = (const float*)d_in[3];   // [3,128,128]
  const float* b1    = (const float*)d_in[4];   // [3,128]
  const float* gamma = (const float*)d_in[5];
  const float* beta  = (const float*)d_in[6];
  const float* W2    = (const float*)d_in[7];
  const float* b2    = (const float*)d_in[8];
  const float* fW1   = (const float*)d_in[9];   // [128,128]
  const float* fb1   = (const float*)d_in[10];
  const float* fW2   = (const float*)d_in[11];  // [128,64]
  const float* fb2   = (const float*)d_in[12];

  const int nE = in_sizes[1] / 2;
  const int ND = N_NODES * DFEAT;

  // Scratch partition (~155 MB).
  float* ws    = (float*)d_ws;
  float* agg   = ws;                        // 100000*128
  float* zb    = agg + (size_t)ND;          // 100000*128
  float* xb    = zb + (size_t)ND;           // 100000*128
  float* g     = xb + (size_t)ND;           // 1024*128
  float* h     = g + (size_t)N_GRAPHS * DFEAT;
  float* stats = h + (size_t)N_GRAPHS * DFEAT;  // 256 floats

  const float invN = 1.0f / (float)N_NODES;
  const int gemm_blocks = N_NODES / 16;     // 6250, exact

  for (int l = 0; l < 3; ++l) {
    const float* X = (l == 0) ? x : xb;
    zero_f32<<<2048, 256, 0, stream>>>(agg, ND);
    zero_f32<<<1, 256, 0, stream>>>(stats, 2 * DFEAT);
    edge_scatter<<<(nE + 7) / 8, 256, 0, stream>>>(X, ei, ei + nE, agg, nE);
    // z1 = (x + agg) @ W1 + b1, plus BN stats accumulation
    gemm_k128<128, 1, 2><<<gemm_blocks, 256, 0, stream>>>(
        X, agg, W1 + (size_t)l * DFEAT * DFEAT, b1 + l * DFEAT, zb, stats,
        nullptr, nullptr, invN);
    // x = relu( relu(BN(z1)) @ W2 + b2 )
    gemm_k128<128, 2, 1><<<gemm_blocks, 256, 0, stream>>>(
        zb, nullptr, W2 + (size_t)l * DFEAT * DFEAT, b2 + l * DFEAT, xb, stats,
        gamma + l * DFEAT, beta + l * DFEAT, invN);
  }

  // global_add_pool
  zero_f32<<<512, 256, 0, stream>>>(g, N_GRAPHS * DFEAT);
  pool_scatter<<<(ND + 255) / 256, 256, 0, stream>>>(xb, batch, g, ND);

  // head MLP
  gemm_k128<128, 0, 1><<<N_GRAPHS / 16, 256, 0, stream>>>(
      g, nullptr, fW1, fb1, h, nullptr, nullptr, nullptr, 0.0f);
  gemm_k128<64, 0, 0><<<N_GRAPHS / 16, 128, 0, stream>>>(
      h, nullptr, fW2, fb2, (float*)d_out, nullptr, nullptr, nullptr, 0.0f);
}